// AttentionPooler_56538949484591
// MI455X (gfx1250) — compile-verified
//
#include <hip/hip_runtime.h>
#include <math.h>

typedef __attribute__((ext_vector_type(2))) float v2f;
typedef __attribute__((ext_vector_type(8))) float v8f;

namespace {
constexpr int kL     = 2048;
constexpr int kD     = 1024;
constexpr int kH     = 256;
constexpr int kS     = 512;
constexpr int kSpan  = 64;
constexpr int kKC    = 32;            // K per staged chunk
constexpr int kChunks = kD / kKC;     // 32
constexpr int kXPit  = kKC + 4;       // 36 floats: 16B-aligned, conflict-free A reads
constexpr int kWPit2 = kH + 16;       // 272 float2: lane-halves hit disjoint bank halves
}

__global__ __launch_bounds__(256, 1)
void attn_pool_span_kernel(const float* __restrict__ x,
                           const int*   __restrict__ spans,
                           const float* __restrict__ Wm,
                           const float* __restrict__ bias,
                           const float* __restrict__ vvec,
                           float*       __restrict__ out)
{
    __shared__ float  Xs[2][kSpan * kXPit];          // 2 x 9 KB
    __shared__ float2 Ws[2][(kKC / 2) * kWPit2];     // 2 x 34 KB, K-pair interleaved
    __shared__ float sh_scores[kSpan];
    __shared__ float sh_att[kSpan];
    __shared__ float sh_pool[kH];

    const int sp   = blockIdx.x;
    const int tid  = threadIdx.x;
    const int lane = tid & 31;
    const int wave = tid >> 5;
    const int half = lane >> 4;      // 0: lanes 0-15, 1: lanes 16-31
    const int l16  = lane & 15;

    const int bi  = spans[3 * sp + 0];
    const int a0  = spans[3 * sp + 1];
    const int len = spans[3 * sp + 2] - a0;   // 1..64, a0+63 < L guaranteed

    if (tid < kSpan) sh_scores[tid] = 0.0f;
    if (tid < kH)    sh_pool[tid]   = 0.0f;

    // 8 waves -> row tile rt (0..3) x column group cg (0..1, 8 col tiles each)
    const int rt = wave & 3;
    const int cg = wave >> 2;

    v8f acc[8];
#pragma unroll
    for (int t = 0; t < 8; ++t)
#pragma unroll
        for (int r = 0; r < 8; ++r) acc[t][r] = 0.0f;

    const size_t xbase = ((size_t)bi * kL + (size_t)a0) * kD;
    const unsigned ldsX[2] = { (unsigned)(uintptr_t)&Xs[0][0],
                               (unsigned)(uintptr_t)&Xs[1][0] };

    float4 wra[4], wrb[4];   // in-flight W tile (regs), 32 VGPRs

    // X tile: 64 rows x 32 floats -> 512 lane-tasks of 16B, async direct to LDS
    auto issue_x = [&](int k0, int buf) {
#pragma unroll
        for (int j = 0; j < 2; ++j) {
            int idx = tid + j * 256;            // 0..511
            int row = idx >> 3;
            int c4  = idx & 7;
            unsigned lds = ldsX[buf] + (unsigned)((row * kXPit + c4 * 4) * 4);
            unsigned long long ga =
                (unsigned long long)(const void*)(x + xbase + (size_t)row * kD + k0 + c4 * 4);
            asm volatile("global_load_async_to_lds_b128 %0, %1, off"
                         :: "v"(lds), "v"(ga) : "memory");
        }
    };
    // W tile: 32 rows x 256 floats; load row pair into regs (for interleave)
    auto issue_w = [&](int k0) {
#pragma unroll
        for (int j = 0; j < 4; ++j) {
            int tt = tid + j * 256;             // 0..1023
            int kp = tt >> 6;                   // K-pair index 0..15
            int c4 = tt & 63;
            const float* p = Wm + (size_t)(k0 + 2 * kp) * kH + c4 * 4;
            wra[j] = *reinterpret_cast<const float4*>(p);
            wrb[j] = *reinterpret_cast<const float4*>(p + kH);
        }
    };
    // interleave {W[k][c], W[k+1][c]} into float2 lanes -> B frag = one ds_load_b64
    auto store_w = [&](int buf) {
#pragma unroll
        for (int j = 0; j < 4; ++j) {
            int tt = tid + j * 256;
            int kp = tt >> 6;
            int c4 = tt & 63;
            float2* q = &Ws[buf][kp * kWPit2 + c4 * 4];
            float4 lo = make_float4(wra[j].x, wrb[j].x, wra[j].y, wrb[j].y);
            float4 hi = make_float4(wra[j].z, wrb[j].z, wra[j].w, wrb[j].w);
            *reinterpret_cast<float4*>(q)     = lo;
            *reinterpret_cast<float4*>(q + 2) = hi;
        }
    };
    auto wait_async = [&]() {
        asm volatile("s_wait_asynccnt 0" ::: "memory");
    };

    auto compute = [&](int buf) {
#pragma unroll
        for (int kk = 0; kk < kKC; kk += 4) {
            // A fragment (16x4 f32): lane half selects K pair {0,1} vs {2,3}
            v2f A;
            {
                float2 af = *reinterpret_cast<const float2*>(
                    &Xs[buf][(rt * 16 + l16) * kXPit + kk + 2 * half]);
                A.x = af.x; A.y = af.y;
            }
            const int krow = (kk >> 1) + half;   // interleaved K-pair row
#pragma unroll
            for (int t = 0; t < 8; ++t) {
                float2 bf = Ws[buf][krow * kWPit2 + (cg * 8 + t) * 16 + l16];
                v2f B; B.x = bf.x; B.y = bf.y;
                acc[t] = __builtin_amdgcn_wmma_f32_16x16x4_f32(
                    false, A, false, B, (short)0, acc[t], false, false);
            }
        }
    };

    // ---- pipelined main loop: async X + reg-staged W double buffering ----
    issue_x(0, 0);
    issue_w(0);
    wait_async();
    store_w(0);
    __syncthreads();

    for (int c = 0; c < kChunks; ++c) {
        const int  cur  = c & 1;
        const bool more = (c + 1 < kChunks);
        if (more) {
            issue_x((c + 1) * kKC, cur ^ 1);
            issue_w((c + 1) * kKC);
        }
        compute(cur);
        if (more) {
            wait_async();
            store_w(cur ^ 1);
        }
        __syncthreads();
    }

    // ---- scores: per-row h . v (bias dropped: softmax shift-invariant) ----
    float rowpart[8];
#pragma unroll
    for (int r = 0; r < 8; ++r) rowpart[r] = 0.0f;
#pragma unroll
    for (int t = 0; t < 8; ++t) {
        const float vv = vvec[(cg * 8 + t) * 16 + l16];
#pragma unroll
        for (int r = 0; r < 8; ++r) rowpart[r] += acc[t][r] * vv;
    }
#pragma unroll
    for (int r = 0; r < 8; ++r) {
        float s = rowpart[r];
        s += __shfl_xor(s, 1, 32);
        s += __shfl_xor(s, 2, 32);
        s += __shfl_xor(s, 4, 32);
        s += __shfl_xor(s, 8, 32);   // sum over the 16 lanes sharing a row
        if (l16 == 0)
            atomicAdd(&sh_scores[rt * 16 + r + 8 * half], s);
    }
    __syncthreads();

    // ---- softmax over valid span positions (<=64 elems, serial is cheap) ----
    if (tid == 0) {
        float m = -INFINITY;
        for (int p = 0; p < len; ++p) m = fmaxf(m, sh_scores[p]);
        float ssum = 0.0f;
        for (int p = 0; p < len; ++p) {
            float e = __expf(sh_scores[p] - m);
            sh_att[p] = e;
            ssum += e;
        }
        float inv = 1.0f / ssum;
        for (int p = 0; p < kSpan; ++p)
            sh_att[p] = (p < len) ? sh_att[p] * inv : 0.0f;
    }
    __syncthreads();

    // ---- pooled = att @ (xW); masked rows have att == 0 ----
#pragma unroll
    for (int t = 0; t < 8; ++t) {
        float p = 0.0f;
#pragma unroll
        for (int r = 0; r < 8; ++r)
            p += sh_att[rt * 16 + r + 8 * half] * acc[t][r];
        atomicAdd(&sh_pool[(cg * 8 + t) * 16 + l16], p);
    }
    __syncthreads();

    // ---- output: pooled + b (bias re-added since sum(att) == 1) ----
    if (tid < kH)
        out[(size_t)sp * kH + tid] = sh_pool[tid] + bias[tid];
}

extern "C" void kernel_launch(void* const* d_in, const int* in_sizes, int n_in,
                              void* d_out, int out_size, void* d_ws, size_t ws_size,
                              hipStream_t stream) {
    const float* x     = (const float*)d_in[0];
    const int*   spans = (const int*)  d_in[1];
    const float* Wm    = (const float*)d_in[2];
    const float* b     = (const float*)d_in[3];
    const float* v     = (const float*)d_in[4];
    float*       outp  = (float*)d_out;
    attn_pool_span_kernel<<<kS, 256, 0, stream>>>(x, spans, Wm, b, v, outp);
}